// MultiHeadAttentionQuantum_65481071399170
// MI455X (gfx1250) — compile-verified
//
#include <hip/hip_runtime.h>

// out = cumprod(cos(x)) over contiguous 64-float segments.
// One wave32 handles 16 segments (4 KB tile). Scan done in log2 domain via
// blocked lower-triangular matmul using exact-fp32 V_WMMA_F32_16X16X4_F32.
// cumprod = exp2(cumsum(log2|cos|)) * (-1)^(cumsum(negbit)), both cumsums
// computed by the same triangular WMMA chain.

typedef __attribute__((ext_vector_type(2))) float v2f;
typedef __attribute__((ext_vector_type(4))) float v4f;
typedef __attribute__((ext_vector_type(8))) float v8f;

__global__ __launch_bounds__(256)
void qcumprod_wmma_kernel(const float* __restrict__ x, float* __restrict__ out) {
  const int lane = threadIdx.x & 31;
  const int wavesPerBlock = blockDim.x >> 5;
  const long long waveId = (long long)blockIdx.x * wavesPerBlock + (threadIdx.x >> 5);

  const float* __restrict__ xt = x + waveId * 1024;
  float*       __restrict__ ot = out + waveId * 1024;

  const int seg = lane & 15;          // segment handled by this lane (N index)
  const int k0  = (lane >> 4) << 1;   // K sub-rows held by this half-wave: {0,1} or {2,3}
  const int m   = seg;                // A-matrix row M = lane % 16

  // A = 16x4 column-chunks of the 16x16 lower-triangular ones matrix.
  // A[q][m][k] = (m >= 4q + k).  Layout: lanes0-15 hold K=0,1; lanes16-31 K=2,3.
  v2f A[4];
#pragma unroll
  for (int q = 0; q < 4; ++q) {
    A[q].x = (m >= 4 * q + k0)     ? 1.0f : 0.0f;
    A[q].y = (m >= 4 * q + k0 + 1) ? 1.0f : 0.0f;
  }

  const float* __restrict__ segp = xt + seg * 64 + k0;                  // B elements
  float*       __restrict__ outp = ot + seg * 64 + ((lane >> 4) << 3);  // 8 contiguous outs

  // Preload the whole tile (16 x float2 per lane) so the 16 global_load_b64
  // issue back-to-back and their latency overlaps the transcendental work.
  v2f v[16];
#pragma unroll
  for (int j = 0; j < 16; ++j)
    v[j] = __builtin_nontemporal_load((const v2f*)(segp + 4 * j));

  v8f accS = {0.f, 0.f, 0.f, 0.f, 0.f, 0.f, 0.f, 0.f};  // cumsum of log2|cos|
  v8f accN = {0.f, 0.f, 0.f, 0.f, 0.f, 0.f, 0.f, 0.f};  // cumsum of sign bits

#pragma unroll
  for (int i = 0; i < 4; ++i) {       // 4 blocks of 16 wires
#pragma unroll
    for (int q = 0; q < 4; ++q) {     // K chunks of 4 wires
      const v2f w = v[4 * i + q];
      // v_cos_f32 takes rotations: multiply by 1/(2*pi) first.
      const float c0 = __builtin_amdgcn_cosf(w.x * 0.15915494309189535f);
      const float c1 = __builtin_amdgcn_cosf(w.y * 0.15915494309189535f);
      v2f bm, bn;
      bm.x = __builtin_amdgcn_logf(fabsf(c0));   // raw v_log_f32 = log2; 0 -> -inf
      bm.y = __builtin_amdgcn_logf(fabsf(c1));
      bn.x = (c0 < 0.0f) ? 1.0f : 0.0f;
      bn.y = (c1 < 0.0f) ? 1.0f : 0.0f;
      // D = A(16x4) x B(4x16) + C  — exact fp32 matrix op
      accS = __builtin_amdgcn_wmma_f32_16x16x4_f32(false, A[q], false, bm,
                                                   (short)0, accS, false, false);
      accN = __builtin_amdgcn_wmma_f32_16x16x4_f32(false, A[q], false, bn,
                                                   (short)0, accN, false, false);
    }

    // Reconstruct cumprod for this 16-wire block: lane holds 8 contiguous wires.
    float r[8];
#pragma unroll
    for (int rr = 0; rr < 8; ++rr) {
      const float val = __builtin_amdgcn_exp2f(accS[rr]);   // raw v_exp_f32 = exp2
      const int par = __float2int_rn(accN[rr]) & 1;         // parity of #negatives
      r[rr] = par ? -val : val;
    }
    v4f o0 = {r[0], r[1], r[2], r[3]};
    v4f o1 = {r[4], r[5], r[6], r[7]};
    __builtin_nontemporal_store(o0, (v4f*)(outp + 16 * i));
    __builtin_nontemporal_store(o1, (v4f*)(outp + 16 * i + 4));

    // Carry = inclusive total through wire 16i+15 = D row M=15 (VGPR7, lanes 16-31).
    if (i < 3) {
      const float cS = __shfl(accS[7], (lane & 15) + 16, 32);
      const float cN = __shfl(accN[7], (lane & 15) + 16, 32);
#pragma unroll
      for (int rr = 0; rr < 8; ++rr) { accS[rr] = cS; accN[rr] = cN; }
    }
  }
}

extern "C" void kernel_launch(void* const* d_in, const int* in_sizes, int n_in,
                              void* d_out, int out_size, void* d_ws, size_t ws_size,
                              hipStream_t stream) {
  (void)in_sizes; (void)n_in; (void)d_ws; (void)ws_size;
  const float* x = (const float*)d_in[0];   // d_in[1] (circuit_params) drops out analytically
  float* out = (float*)d_out;

  // 1024 floats (16 segments) per wave, 8 waves per 256-thread block.
  const long long waves  = (long long)out_size / 1024;   // 32768 for 8x4096x1024
  const int       blocks = (int)((waves + 7) / 8);
  qcumprod_wmma_kernel<<<blocks, 256, 0, stream>>>(x, out);
}